// ContrastMemory_83116207112570
// MI455X (gfx1250) — compile-verified
//
#include <hip/hip_runtime.h>
#include <math.h>

typedef __attribute__((ext_vector_type(2))) float v2f;
typedef __attribute__((ext_vector_type(4))) float v4f;
typedef __attribute__((ext_vector_type(8))) float v8f;

#define FEAT_DIM   128
#define N_DATA_SZ  1000000
#define KSZ        8193            // K+1
#define BSZ        64
#define T_INV      (1.0f / 0.07f)
#define MOM        0.5f
#define EPSV       1e-12f

#define TILES_PER_B   513                       // ceil(8193/16)
#define WAVES_TOTAL   (BSZ * TILES_PER_B)       // 32832
#define WAVES_PER_BLK 8
#define NBLK          (WAVES_TOTAL / WAVES_PER_BLK)  // 4104 (exact)
#define OUT_STRIDE    (BSZ * KSZ)               // 524352
#define WS_PART_OFF   16                        // ws[0..1] = inv_z; partials after

// One wave: 16 k-rows of one batch element b. A = gathered memory rows (16x4
// f32 tile per step), B = feat chunk broadcast across all 16 columns, chained
// D += A*B over 32 steps covering d=0..127. Lanes 0..15 end with rows 0..7 in
// c[0..7]; lanes 16..31 with rows 8..15 (per ISA 16x16 f32 C/D layout).
__global__ __launch_bounds__(256) void contrast_kernel(
    const float* __restrict__ mem, const float* __restrict__ feat,
    const int* __restrict__ sample_idx, float* __restrict__ out,
    float* __restrict__ partial) {
  __shared__ float wsum[WAVES_PER_BLK];
  const int tid  = threadIdx.x;
  const int lane = tid & 31;
  const int wave = tid >> 5;
  const int task = blockIdx.x * WAVES_PER_BLK + wave;   // exact fit, no tail
  const int b    = task / TILES_PER_B;
  const int kb   = (task % TILES_PER_B) * 16;

  const int r  = lane & 15;          // k-row within tile for this lane
  const int hi = lane >> 4;          // 0 -> K pair {0,1}, 1 -> {2,3}
  int kk = kb + r;
  if (kk > KSZ - 1) kk = KSZ - 1;    // clamp for addressing; masked later
  const long long row = (long long)sample_idx[b * KSZ + kk];
  const float* rp = mem + row * FEAT_DIM + hi * 2;
  const float* fp = feat + (long long)b * FEAT_DIM + hi * 2;

  v8f c = {};
#pragma unroll
  for (int t = 0; t < 32; ++t) {
    v2f a  = *(const v2f*)(rp + 4 * t);
    v2f bb = *(const v2f*)(fp + 4 * t);
    c = __builtin_amdgcn_wmma_f32_16x16x4_f32(false, a, false, bb,
                                              (short)0, c, false, false);
  }

  float ev[8];
  float s = 0.0f;
#pragma unroll
  for (int j = 0; j < 8; ++j) {
    const int m = hi * 8 + j;
    const float e = expf(fminf(c[j] * T_INV, 50.0f));
    ev[j] = e;
    if (kb + m < KSZ) s += e;
  }
  if (r == 0) {  // lanes 0 and 16 own unique rows; store unnormalized exp
#pragma unroll
    for (int j = 0; j < 8; ++j) {
      const int m = hi * 8 + j;
      if (kb + m < KSZ) out[(long long)b * KSZ + kb + m] = ev[j];
    }
  }
  // deterministic wave reduce: only lanes 0 and 16 contribute
  s = (r == 0) ? s : 0.0f;
  s += __shfl_xor(s, 16, 32);
  if (lane == 0) wsum[wave] = s;
  __syncthreads();
  if (tid == 0) {
    float bs = 0.0f;
#pragma unroll
    for (int w = 0; w < WAVES_PER_BLK; ++w) bs += wsum[w];
    partial[blockIdx.x] = bs;   // fixed slot -> deterministic
  }
}

__global__ void reduce_kernel(float* __restrict__ ws) {
  const int c = threadIdx.x;
  if (c < 2) {
    const float* p = ws + WS_PART_OFF + (long long)c * NBLK;
    float s = 0.0f;
    for (int i = 0; i < NBLK; ++i) s += p[i];   // fixed order
    const float z = (s / (float)(BSZ * KSZ)) * (float)N_DATA_SZ;
    ws[c] = 1.0f / z;
  }
}

__global__ __launch_bounds__(256) void scale_kernel(float* __restrict__ out,
                                                    const float* __restrict__ ws) {
  const int i = blockIdx.x * 256 + threadIdx.x;
  if (i < 2 * OUT_STRIDE) {
    const float inv = ws[(i >= OUT_STRIDE) ? 1 : 0];
    out[i] *= inv;
  }
}

__global__ __launch_bounds__(256) void copy_mem_kernel(
    const float* __restrict__ src, float* __restrict__ dst) {
  const v4f* s = (const v4f*)src;
  v4f* d = (v4f*)dst;
  const long long n = (long long)N_DATA_SZ * FEAT_DIM / 4;  // 32,000,000
  long long i = (long long)blockIdx.x * blockDim.x + threadIdx.x;
  const long long stride = (long long)gridDim.x * blockDim.x;
  for (; i < n; i += stride) d[i] = s[i];
}

// One wave per (bank, batch row): pos = 0.5*old + 0.5*feat, L2-normalize.
__global__ __launch_bounds__(256) void update_rows_kernel(
    const float* __restrict__ mem_s_old, const float* __restrict__ mem_t_old,
    const float* __restrict__ feat_s, const float* __restrict__ feat_t,
    const int* __restrict__ idx, float* __restrict__ mem_s_new,
    float* __restrict__ mem_t_new) {
  const int g    = (blockIdx.x * blockDim.x + threadIdx.x) >> 5;  // global wave
  const int lane = threadIdx.x & 31;
  if (g >= 2 * BSZ) return;
  const int bank = g >> 6;  // 0 = s, 1 = t
  const int b    = g & 63;
  const float* mo = bank ? mem_t_old : mem_s_old;
  const float* ft = bank ? feat_t : feat_s;
  float* mn       = bank ? mem_t_new : mem_s_new;
  const long long row = (long long)idx[b];
  const float* rp = mo + row * FEAT_DIM;
  const float* fp = ft + (long long)b * FEAT_DIM;
  float p[4];
  float ss = 0.0f;
#pragma unroll
  for (int j = 0; j < 4; ++j) {
    const float v = rp[lane * 4 + j] * MOM + fp[lane * 4 + j] * (1.0f - MOM);
    p[j] = v;
    ss += v * v;
  }
#pragma unroll
  for (int off = 16; off > 0; off >>= 1) ss += __shfl_xor(ss, off, 32);
  const float inv = 1.0f / fmaxf(sqrtf(ss), EPSV);
  float* op = mn + row * FEAT_DIM;
#pragma unroll
  for (int j = 0; j < 4; ++j) op[lane * 4 + j] = p[j] * inv;
}

extern "C" void kernel_launch(void* const* d_in, const int* in_sizes, int n_in,
                              void* d_out, int out_size, void* d_ws,
                              size_t ws_size, hipStream_t stream) {
  (void)in_sizes; (void)n_in; (void)out_size; (void)ws_size;
  const float* feat_s     = (const float*)d_in[0];
  const float* feat_t     = (const float*)d_in[1];
  const int*   idx        = (const int*)d_in[2];
  const int*   sample_idx = (const int*)d_in[3];
  const float* memory_s   = (const float*)d_in[4];
  const float* memory_t   = (const float*)d_in[5];

  float* out = (float*)d_out;
  float* ws  = (float*)d_ws;

  float* out_s     = out;                          // [64, 8193]
  float* out_t     = out + OUT_STRIDE;             // [64, 8193]
  float* mem_s_new = out + 2LL * OUT_STRIDE;       // [1e6, 128]
  float* mem_t_new = mem_s_new + (long long)N_DATA_SZ * FEAT_DIM;

  // out_s = contrast(memory_t, feat_s); out_t = contrast(memory_s, feat_t)
  contrast_kernel<<<NBLK, 256, 0, stream>>>(memory_t, feat_s, sample_idx,
                                            out_s, ws + WS_PART_OFF);
  contrast_kernel<<<NBLK, 256, 0, stream>>>(memory_s, feat_t, sample_idx,
                                            out_t, ws + WS_PART_OFF + NBLK);
  reduce_kernel<<<1, 32, 0, stream>>>(ws);
  scale_kernel<<<(2 * OUT_STRIDE + 255) / 256, 256, 0, stream>>>(out, ws);

  copy_mem_kernel<<<8192, 256, 0, stream>>>(memory_s, mem_s_new);
  copy_mem_kernel<<<8192, 256, 0, stream>>>(memory_t, mem_t_new);
  update_rows_kernel<<<16, 256, 0, stream>>>(memory_s, memory_t, feat_s,
                                             feat_t, idx, mem_s_new, mem_t_new);
}